// roiPooling_50431505989745
// MI455X (gfx1250) — compile-verified
//
#include <hip/hip_runtime.h>

// RoIAlign-style crop+bilinear-resize, tuned for MI455X (gfx1250).
//
// Roofline: output = 1024*14*14*512 f32 = 411 MB -> ~17.6 us at 23.3 TB/s HBM.
// Input image is 8 MB (fully resident in the 192 MB L2); the 4-point gather is
// cache traffic, not HBM traffic. Total arithmetic ~0.8 GFLOP -> the kernel is
// purely store-bandwidth-bound. Design:
//   - One block per (roi, py) output row; the 14 px positions are fully
//     unrolled -> the gfx1250 workgroup-ID preamble, ROI s_load and all y-side
//     scalar-float math (s_fmaak_f32 / s_cvt_f32_i32) amortize over 14 stores,
//     and each wave keeps many b128 loads + NT stores in flight.
//   - Half-pixel fractions (p+0.5)/14 as compile-time-folded constants
//     (bit-identical to IEEE runtime division).
//   - Lanes only do coalesced global_load_b128 gathers + lerps + non-temporal
//     global_store_b128 (TH_STORE_NT) so the write-once 411 MB stream never
//     evicts the L2-resident image.

#define POOL_P    14
#define IMG_H     64
#define IMG_W     64
#define IMG_C     512
#define C4        (IMG_C / 4)     // 128 float4 per pixel

typedef float v4f __attribute__((ext_vector_type(4)));

// (p + 0.5f) / 14.0f, folded at compile time with IEEE f32 rounding ->
// identical to the reference's on-device division.
__device__ __constant__ float kGrid[POOL_P] = {
    0.5f / 14.0f,  1.5f / 14.0f,  2.5f / 14.0f,  3.5f / 14.0f,
    4.5f / 14.0f,  5.5f / 14.0f,  6.5f / 14.0f,  7.5f / 14.0f,
    8.5f / 14.0f,  9.5f / 14.0f, 10.5f / 14.0f, 11.5f / 14.0f,
   12.5f / 14.0f, 13.5f / 14.0f
};

static __device__ __forceinline__ v4f lerp4(v4f a, v4f b, float t) {
    const float u = 1.0f - t;
    v4f r;
    r.x = a.x * u + b.x * t;
    r.y = a.y * u + b.y * t;
    r.z = a.z * u + b.z * t;
    r.w = a.w * u + b.w * t;
    return r;
}

// grid = (py:14, roi:1024); 128 threads = 4 wave32s over the channel dim as
// float4s; each block emits one full 14-px output row.
__global__ __launch_bounds__(128) void roi_align_row_kernel(
    const float* __restrict__ img,   // [64, 64, 512]
    const int*   __restrict__ rois,  // [1024, 4] (x, y, w, h)
    float*       __restrict__ out)   // [1024, 14, 14, 512]
{
    const int py  = blockIdx.x;
    const int roi = blockIdx.y;

    // Block-uniform ROI box -> scalar load.
    const int rx = rois[roi * 4 + 0];
    const int ry = rois[roi * 4 + 1];
    const int rw = rois[roi * 4 + 2];
    const int rh = rois[roi * 4 + 3];

    // TF2 half-pixel coords (matches reference):
    //   src = grid[p] * size - 0.5 ; lo/hi clamped to [0, size-1];
    //   t = src - floor(src) (unclamped; harmless when lo == hi).
    const float sy = kGrid[py] * (float)rh - 0.5f;
    const float fy = floorf(sy);
    const float ty = sy - fy;
    const int   iy = (int)fy;
    const int   ylo = ry + min(max(iy,     0), rh - 1);
    const int   yhi = ry + min(max(iy + 1, 0), rh - 1);

    const v4f* __restrict__ imgv = (const v4f*)img;
    v4f* outv = (v4f*)out;
    const int c4 = threadIdx.x;

    const int rowLo = ylo * IMG_W;   // block-uniform row bases (pixel units)
    const int rowHi = yhi * IMG_W;
    const long long outRow =
        (((long long)roi * POOL_P + py) * POOL_P) * C4 + c4;

#pragma unroll
    for (int px = 0; px < POOL_P; ++px) {
        const float sx = kGrid[px] * (float)rw - 0.5f;
        const float fx = floorf(sx);
        const float tx = sx - fx;
        const int   ix = (int)fx;
        const int   xlo = rx + min(max(ix,     0), rw - 1);
        const int   xhi = rx + min(max(ix + 1, 0), rw - 1);

        const v4f v00 = imgv[(rowLo + xlo) * C4 + c4];
        const v4f v01 = imgv[(rowLo + xhi) * C4 + c4];
        const v4f v10 = imgv[(rowHi + xlo) * C4 + c4];
        const v4f v11 = imgv[(rowHi + xhi) * C4 + c4];

        const v4f top = lerp4(v00, v01, tx);
        const v4f bot = lerp4(v10, v11, tx);
        const v4f o   = lerp4(top, bot, ty);

        // Streamed, write-once output: non-temporal 128-bit store.
        __builtin_nontemporal_store(o, &outv[outRow + (long long)px * C4]);
    }
}

extern "C" void kernel_launch(void* const* d_in, const int* in_sizes, int n_in,
                              void* d_out, int out_size, void* d_ws, size_t ws_size,
                              hipStream_t stream) {
    (void)in_sizes; (void)n_in; (void)out_size; (void)d_ws; (void)ws_size;
    const float* img  = (const float*)d_in[0];  // [1,64,64,512] f32
    const int*   rois = (const int*)d_in[1];    // [1,1024,4]   i32
    float*       out  = (float*)d_out;          // [1,1024,14,14,512] f32

    dim3 grid(POOL_P, 1024);                    // (py, roi)
    roi_align_row_kernel<<<grid, 128, 0, stream>>>(img, rois, out);
}